// TrainShiftInvariantLoss_2645699854801
// MI455X (gfx1250) — compile-verified
//
#include <hip/hip_runtime.h>
#include <hip/hip_bf16.h>

// Shift-invariant BCE loss for MI455X (gfx1250).
//
// Key identity: BCE elem = max(l,0) - l*t + log1p(exp(-|l|)).  Only the -l*t
// term depends on the shifted target window, so
//   errs[k,b] = (F[b] - C[b,k]) / (h*w)
// with F[b] shift-invariant and C[b,k] a 17x17 cross-correlation of the
// 224x224 logits against the 240x240 targets.  argmin errs == argmax C.
// The correlation is expressed as an implicit GEMM on V_WMMA_F32_16X16X4_F32
// (fp32 throughout to preserve argmin fidelity):
//   k=(x,y'), A[i,k]=T[x+i,y'], B[k,j]=L[x,y'-j]  =>  D[i,j]=C[i,j].

typedef __attribute__((ext_vector_type(2))) float v2f;
typedef __attribute__((ext_vector_type(8))) float v8f;

#define NB     8
#define HIN    224
#define WIN    224
#define HT     240
#define WT     240
#define SSH    17          // shifts per dim
#define NSHIFT (SSH*SSH)   // 289
#define HWIN   (HIN*WIN)   // 50176
#define HWT    (HT*WT)     // 57600
#define NSLICE 14
#define ROWS_PER_SLICE (HIN / NSLICE)   // 16

// ---------------- kernel 0: zero the scratch accumulators + out[0] ---------
__global__ void ws_zero_kernel(float* ws, float* out0) {
    int i = blockIdx.x * blockDim.x + threadIdx.x;
    if (i < NB * NSHIFT + NB) ws[i] = 0.0f;   // C[8][289] then F[8]
    if (i == 0) out0[0] = 0.0f;
}

// ---------------- kernel 1: F[b] = sum(max(l,0) + log1p(exp(-|l|))) --------
__global__ void f_reduce_kernel(const float* __restrict__ L, float* __restrict__ F) {
    const int b = blockIdx.x;
    const float* Lb = L + b * HWIN;
    float s = 0.0f;
    for (int idx = threadIdx.x; idx < HWIN; idx += blockDim.x) {
        float l = Lb[idx];
        s += fmaxf(l, 0.0f) + log1pf(expf(-fabsf(l)));
    }
    #pragma unroll
    for (int off = 16; off > 0; off >>= 1) s += __shfl_down(s, off, 32);
    if ((threadIdx.x & 31) == 0) atomicAdd(&F[b], s);
}

// ---------------- kernel 2: WMMA cross-correlation -------------------------
// grid.x = NB*NSLICE, block = 128 (4 waves).  Wave w handles tile (mt,nt).
__global__ __launch_bounds__(128)
void corr_wmma_kernel(const float* __restrict__ L, const float* __restrict__ T,
                      float* __restrict__ C) {
    const int b     = blockIdx.x / NSLICE;
    const int slice = blockIdx.x % NSLICE;
    const int wave  = threadIdx.x >> 5;
    const int lane  = threadIdx.x & 31;
    const int mt = wave >> 1, nt = wave & 1;
    const int Moff = mt * 16, Noff = nt * 16;
    const int half = lane >> 4;       // K-half within chunk of 4
    const int m    = lane & 15;       // A row index / B column index

    const float* Lb = L + b * HWIN;
    const float* Tb = T + b * HWT;

    // A-operand source row of T (clamped; clamped rows feed only discarded
    // D rows with global i >= 17).
    int trowOff = Moff + m;

    v8f acc = {};
    const int x0 = slice * ROWS_PER_SLICE;
    for (int x = x0; x < x0 + ROWS_PER_SLICE; ++x) {
        const float* Lrow = Lb + x * WIN;
        int trow = x + trowOff;
        trow = (trow > HT - 1) ? (HT - 1) : trow;
        const float* Trow = Tb + trow * WT;
        for (int y0 = 0; y0 < WT; y0 += 4) {
            const int ya = y0 + 2 * half;            // y' for VGPR0 of this lane
            // A: lane (half,m) holds A[m, k0+2*half] and A[m, k0+2*half+1]
            v2f a;
            a.x = Trow[ya];
            a.y = Trow[ya + 1];
            // B: lane (half,m) holds B[k0+2*half, m] and B[k0+2*half+1, m]
            //    B[k,j] = L[x, y' - (Noff+j)], zero outside [0,224)
            const int c0 = ya - (Noff + m);
            const int c1 = c0 + 1;
            int c0c = c0 < 0 ? 0 : (c0 > WIN - 1 ? WIN - 1 : c0);
            int c1c = c1 < 0 ? 0 : (c1 > WIN - 1 ? WIN - 1 : c1);
            float b0 = Lrow[c0c];
            float b1 = Lrow[c1c];
            b0 = (c0 >= 0 && c0 < WIN) ? b0 : 0.0f;  // v_cndmask, wave-uniform CF
            b1 = (c1 >= 0 && c1 < WIN) ? b1 : 0.0f;
            v2f bv; bv.x = b0; bv.y = b1;
            // D = A x B + C   (f32 16x16x4)
            acc = __builtin_amdgcn_wmma_f32_16x16x4_f32(
                false, a, false, bv, (short)0, acc, false, false);
        }
    }

    // C/D layout: VGPR v -> (M = v + 8*half, N = m).  Keep only i,j < 17.
    #pragma unroll
    for (int v = 0; v < 8; ++v) {
        int i = Moff + v + 8 * half;
        int j = Noff + m;
        if (i < SSH && j < SSH)
            atomicAdd(&C[b * NSHIFT + i * SSH + j], acc[v]);
    }
}

// ---------------- kernel 3: per-sample argmax + total loss -----------------
__global__ void argmax_loss_kernel(const float* __restrict__ C,
                                   const float* __restrict__ F,
                                   float* __restrict__ out0,
                                   int* __restrict__ bestK) {
    const int b = threadIdx.x >> 5;      // 8 waves, one sample each
    const int lane = threadIdx.x & 31;
    float best = -__builtin_inff();
    int bk = 0x7fffffff;
    for (int k = lane; k < NSHIFT; k += 32) {
        float v = C[b * NSHIFT + k];
        if (v > best) { best = v; bk = k; }   // increasing k => first-max kept
    }
    #pragma unroll
    for (int off = 16; off > 0; off >>= 1) {
        float ov = __shfl_down(best, off, 32);
        int   ok = __shfl_down(bk, off, 32);
        if (ov > best || (ov == best && ok < bk)) { best = ov; bk = ok; }
    }
    if (lane == 0) {
        bestK[b] = bk;
        float lossb = (F[b] - best) * (1.0f / (float)HWIN);
        atomicAdd(out0, lossb);
    }
}

// ---------------- kernel 4: gather the winning window ----------------------
__global__ void gather_kernel(const float* __restrict__ T,
                              const int* __restrict__ bestK,
                              float* __restrict__ out) {
    int idx = blockIdx.x * blockDim.x + threadIdx.x;
    if (idx >= NB * HWIN) return;
    int b = idx / HWIN;
    int r = idx - b * HWIN;
    int x = r / WIN, y = r - x * WIN;
    int k = bestK[b];
    int i = k / SSH, j = k - i * SSH;
    out[1 + idx] = T[b * HWT + (x + i) * WT + (y + j)];
}

extern "C" void kernel_launch(void* const* d_in, const int* in_sizes, int n_in,
                              void* d_out, int out_size, void* d_ws, size_t ws_size,
                              hipStream_t stream) {
    const float* L = (const float*)d_in[0];   // [8,1,224,224] logits
    const float* T = (const float*)d_in[1];   // [8,1,240,240] targets
    float* out = (float*)d_out;               // [0]=loss, [1..]=adjusted

    // workspace: C[8*289] | F[8] | bestK[8] (int)
    float* Cws = (float*)d_ws;
    float* Fws = Cws + NB * NSHIFT;
    int* bestK = (int*)(Fws + NB);

    ws_zero_kernel<<<(NB * NSHIFT + NB + 255) / 256, 256, 0, stream>>>(Cws, out);
    f_reduce_kernel<<<NB, 256, 0, stream>>>(L, Fws);
    corr_wmma_kernel<<<NB * NSLICE, 128, 0, stream>>>(L, T, Cws);
    argmax_loss_kernel<<<1, NB * 32, 0, stream>>>(Cws, Fws, out, bestK);
    gather_kernel<<<(NB * HWIN + 255) / 256, 256, 0, stream>>>(T, bestK, out);
}